// PreprocessingLayer_16475494547869
// MI455X (gfx1250) — compile-verified
//
#include <hip/hip_runtime.h>
#include <cmath>

typedef __attribute__((ext_vector_type(2))) float v2f;
typedef __attribute__((ext_vector_type(8))) float v8f;

#define RADIUS   14
#define TAPS     29          // 2*R+1
#define TILE     64
#define HALO     (TILE + 2*RADIUS)   // 92
// LDS row pitch (floats). 108 mod 64 = 44, gcd(44,64)=4 -> 16 consecutive rows
// hit 16 distinct banks (conflict-free column access in stage 2); 2*108 mod 64
// = 24 and 8*108 mod 64 = 32 keep the two lane-halves' banks disjoint for the
// row-pair reads (stage 1) and the C/D row-half stores. 96 would give 8-way
// conflicts on stage-2 A loads (32*r mod 64 spans only 2 banks).
#define PITCH    108
#define IMG      512

struct GaussParams {
    float w[TAPS];   // 1-D gaussian taps
    float M[9];      // 3x3 channel-mix (symmetric-reflect blur of length-3 axis)
};

__host__ __device__ inline int sym_index(int j, int L) {
    int p = 2 * L;
    j %= p; if (j < 0) j += p;
    return (j < L) ? j : (p - 1 - j);
}

// Fused gaussian blur: channel mix at load -> H blur (WMMA) -> W blur (WMMA).
// One 64x64x3 output tile per workgroup; 92x92x3 mixed input region in LDS.
__global__ void __launch_bounds__(256)
gauss_blur_fused_wmma(const float* __restrict__ img, float* __restrict__ out,
                      GaussParams gp)
{
    extern __shared__ float lds[];
    float* lds_w   = lds;                          // 32 floats
    float* lds_in  = lds + 32;                     // 3 * 92 * 108
    float* lds_mid = lds_in + 3 * HALO * PITCH;    // 64 * 108

    const int tid  = threadIdx.x;
    const int lane = tid & 31;
    const int wave = tid >> 5;
    const int b    = blockIdx.z;
    const int gr0  = blockIdx.y * TILE;
    const int gc0  = blockIdx.x * TILE;

    if (tid < TAPS) lds_w[tid] = gp.w[tid];

    // ---- stage 0: global load + 3x3 channel mix into channel-planar LDS ----
    for (int i = tid; i < HALO * HALO; i += 256) {
        const int r  = i / HALO, c = i % HALO;
        const int gy = sym_index(gr0 - RADIUS + r, IMG);
        const int gx = sym_index(gc0 - RADIUS + c, IMG);
        const float* p = img + (((long)b * IMG + gy) * IMG + gx) * 3;
        const float x0 = p[0], x1 = p[1], x2 = p[2];
        lds_in[(0 * HALO + r) * PITCH + c] = gp.M[0]*x0 + gp.M[1]*x1 + gp.M[2]*x2;
        lds_in[(1 * HALO + r) * PITCH + c] = gp.M[3]*x0 + gp.M[4]*x1 + gp.M[5]*x2;
        lds_in[(2 * HALO + r) * PITCH + c] = gp.M[6]*x0 + gp.M[7]*x1 + gp.M[8]*x2;
    }
    __syncthreads();

    // ---- per-lane Toeplitz weight operand (shared by stage-1 A and stage-2 B)
    // A layout (ISA): VGPR0 = K{0,2}, VGPR1 = K{1,3}; lanes0-15 M, lanes16-31 M.
    // B layout mirrors C/D striping: VGPR j = rows {j, j+2} across lane halves.
    // Both reduce to w[k0 + kb + {0,1} - (lane&15)].
    const int m  = lane & 15;
    const int kb = (lane >> 4) * 2;   // 0 or 2
    v2f wt[11];
#pragma unroll
    for (int kc = 0; kc < 11; ++kc) {
        const int i0 = 4 * kc + kb - m, i1 = i0 + 1;
        wt[kc].x = (i0 >= 0 && i0 < TAPS) ? lds_w[i0] : 0.0f;
        wt[kc].y = (i1 >= 0 && i1 < TAPS) ? lds_w[i1] : 0.0f;
    }

    for (int ch = 0; ch < 3; ++ch) {
        const float* inp = lds_in + ch * HALO * PITCH;

        // ---- stage 1: H blur. D[m][n] = sum_j w[j-m] * in[rt+j][ct+n] ----
        // 4 x 6 = 24 tiles of 16x16 over (64 rows x 96 cols); 3 tiles/wave.
        for (int t = wave; t < 24; t += 8) {
            const int rt = (t / 6) * 16, ct = (t % 6) * 16;
            v8f acc = {};
#pragma unroll
            for (int kc = 0; kc < 11; ++kc) {
                const int k0 = 4 * kc;
                v2f bmat;   // input rows rt+k0+{kb, kb+1}, col ct+m
                bmat.x = inp[(rt + k0 + kb + 0) * PITCH + ct + m];
                bmat.y = inp[(rt + k0 + kb + 1) * PITCH + ct + m];
                acc = __builtin_amdgcn_wmma_f32_16x16x4_f32(
                        false, wt[kc], false, bmat, (short)0, acc, false, false);
            }
            const int row0 = rt + ((lane >> 4) << 3);   // C/D: M = v (+8 hi half)
#pragma unroll
            for (int v = 0; v < 8; ++v)
                lds_mid[(row0 + v) * PITCH + ct + m] = acc[v];
        }
        __syncthreads();

        // ---- stage 2: W blur. D[m][n] = sum_j mid[rt+m][ct+j] * w[j-n] ----
        // 4 x 4 = 16 tiles over 64x64; 2 tiles/wave.
        for (int t = wave; t < 16; t += 8) {
            const int rt = (t / 4) * 16, ct = (t % 4) * 16;
            v8f acc = {};
#pragma unroll
            for (int kc = 0; kc < 11; ++kc) {
                const int k0 = 4 * kc;
                // mid row rt+m, adjacent cols ct+k0+kb, +1: 8-byte aligned
                // (every term even) -> single ds_load_b64.
                const v2f amat =
                    *(const v2f*)&lds_mid[(rt + m) * PITCH + ct + k0 + kb];
                acc = __builtin_amdgcn_wmma_f32_16x16x4_f32(
                        false, amat, false, wt[kc], (short)0, acc, false, false);
            }
            const int row0 = gr0 + rt + ((lane >> 4) << 3);
            const int col  = gc0 + ct + m;
#pragma unroll
            for (int v = 0; v < 8; ++v)
                out[(((long)b * IMG + row0 + v) * IMG + col) * 3 + ch] = acc[v];
        }
        __syncthreads();
    }
}

extern "C" void kernel_launch(void* const* d_in, const int* in_sizes, int n_in,
                              void* d_out, int out_size, void* d_ws, size_t ws_size,
                              hipStream_t stream) {
    const float* img = (const float*)d_in[0];
    float* out = (float*)d_out;
    const int B = in_sizes[0] / (IMG * IMG * 3);

    // Host-side constant prep (deterministic, matches reference f64 weights).
    GaussParams gp;
    double wd[TAPS], s = 0.0;
    const double sigma = 3.5;
    for (int i = 0; i < TAPS; ++i) {
        const double x = (double)(i - RADIUS);
        wd[i] = std::exp(-0.5 * (x / sigma) * (x / sigma));
        s += wd[i];
    }
    for (int i = 0; i < TAPS; ++i) gp.w[i] = (float)(wd[i] / s);
    // Channel-axis blur (L=3, symmetric extension) folds into a 3x3 matrix.
    for (int c = 0; c < 3; ++c) {
        double acc[3] = {0.0, 0.0, 0.0};
        for (int k = 0; k < TAPS; ++k)
            acc[sym_index(c - RADIUS + k, 3)] += wd[k] / s;
        for (int ci = 0; ci < 3; ++ci) gp.M[c * 3 + ci] = (float)acc[ci];
    }

    const dim3 grid(IMG / TILE, IMG / TILE, B);   // 8 x 8 x 64
    const size_t lds_bytes =
        (size_t)(32 + 3 * HALO * PITCH + TILE * PITCH) * sizeof(float); // ~147 KB
    gauss_blur_fused_wmma<<<grid, 256, lds_bytes, stream>>>(img, out, gp);
}